// AttnGRUDecoder_63367947485283
// MI455X (gfx1250) — compile-verified
//
#include <hip/hip_runtime.h>

// ---------------------------------------------------------------------------
// AttnGRU greedy decoder for MI455X (gfx1250, wave32).
// GEMMs use V_WMMA_F32_16X16X4_F32 (f32 kept to preserve argmax feedback).
// Weight set (~186MB) fits the 192MB L2 -> loop runs at L2 bandwidth, so the
// GEMM mapping keeps the weight (B) stream single-pass: each wave owns MT
// M-tiles for one N-tile, one B-fragment load feeds MT WMMAs, fragments are
// loaded as b128 (K-permuted) and software-pipelined one chunk ahead.
// ---------------------------------------------------------------------------

#define HDIM 1024
#define VDIM 32000
#define BDIM 64
#define SDIM 100

typedef float v2f __attribute__((ext_vector_type(2)));
typedef float v4f __attribute__((ext_vector_type(4)));
typedef float v8f __attribute__((ext_vector_type(8)));

// ------------------------------ reductions (wave32) ------------------------
__device__ __forceinline__ float wave_sum(float v) {
    #pragma unroll
    for (int off = 16; off > 0; off >>= 1) v += __shfl_down(v, off, 32);
    return v;
}
__device__ __forceinline__ float wave_max(float v) {
    #pragma unroll
    for (int off = 16; off > 0; off >>= 1) v = fmaxf(v, __shfl_down(v, off, 32));
    return v;
}
__device__ __forceinline__ void wave_argmax(float& v, int& i) {
    #pragma unroll
    for (int off = 16; off > 0; off >>= 1) {
        float ov = __shfl_down(v, off, 32);
        int   oi = __shfl_down(i, off, 32);
        if (ov > v || (ov == v && oi < i)) { v = ov; i = oi; }
    }
}
__device__ __forceinline__ float sigmoidf_(float x) { return 1.0f / (1.0f + __expf(-x)); }

__device__ __forceinline__ v2f lo2(v4f x) { return __builtin_shufflevector(x, x, 0, 1); }
__device__ __forceinline__ v2f hi2(v4f x) { return __builtin_shufflevector(x, x, 2, 3); }

// ------------------------------ WMMA GEMM ----------------------------------
// C[m,n] = sum_k A[m,k] * Bt[n,k] + bias[n]
// A: [M,K] row-major (lda), Bt: [N,K] row-major (ldb), C: [M,N] (ldc, 64-bit).
//
// Each wave: MT vertical 16x16 tiles (one N-tile, MT M-tiles); 4 waves/block
// along N.  M % (16*MT) == 0, N % 16 == 0, K % 8 == 0 (all call sites comply).
//
// K-chunks of 8 use a position permutation so ONE b128 load per matrix feeds
// TWO WMMA fragments (WMMA position p only requires A and B to agree on K(p)):
//   lane half hl loads  row[k + 4*hl .. k + 4*hl + 3]  (b128, 16B aligned)
//   frag0: positions(0,1)->K{k,k+1},   (2,3)->K{k+4,k+5}  = lo halves
//   frag1: positions(0,1)->K{k+2,k+3}, (2,3)->K{k+6,k+7}  = hi halves
// The b128 result's even-aligned v2f halves alias WMMA sources directly.
// C/D layout: vgpr j -> row m0 + j + 8*hl, col n0 + (lane & 15).
//
// unroll 1 is deliberate: it preserves the distance-1 rotated register sets
// (loads for chunk c+1 in flight across the 8 WMMAs of chunk c, staggered
// s_wait_loadcnt 3..0); larger unrolls let the scheduler collapse the
// rotation into one reused quad and serialize on s_wait_loadcnt 0.
template <int MT>
__global__ void gemm_atbt_kernel(const float* __restrict__ A, int lda,
                                 const float* __restrict__ Bt, int ldb,
                                 const float* __restrict__ bias,
                                 float* __restrict__ C, long long ldc,
                                 int N, int K)
{
    const int lane = threadIdx.x & 31;
    const int wave = threadIdx.x >> 5;          // 0..3
    const int hl   = lane >> 4;                 // 0 or 1
    const int lr   = lane & 15;

    const int tileN = blockIdx.x * 4 + wave;
    if (tileN * 16 >= N) return;                // wave-uniform; EXEC stays all-1s
    const int m0 = blockIdx.y * (16 * MT);
    const int n0 = tileN * 16;

    const float* Ap = A  + (size_t)(m0 + lr) * (size_t)lda + 4 * hl;
    const float* Bp = Bt + (size_t)(n0 + lr) * (size_t)ldb + 4 * hl;
    const size_t ArowStep = (size_t)16 * (size_t)lda;   // next M-tile

    v8f acc[MT];
    #pragma unroll
    for (int i = 0; i < MT; ++i) acc[i] = (v8f){};

    const int nch = K >> 3;                     // chunks of 8 K-values

    // pipeline stage 0
    v4f bc = *(const v4f*)(Bp);
    v4f ac[MT];
    #pragma unroll
    for (int i = 0; i < MT; ++i) ac[i] = *(const v4f*)(Ap + (size_t)i * ArowStep);

    #pragma unroll 1
    for (int c = 0; c < nch - 1; ++c) {
        const size_t koff = (size_t)(c + 1) << 3;
        // stream-ahead hint on the weight stream (global_prefetch_b8);
        // speculative OOB prefetch is silently dropped.
        __builtin_prefetch(Bp + koff + 1024, 0, 1);
        // preload chunk c+1 while chunk c feeds the matrix pipe
        v4f bn = *(const v4f*)(Bp + koff);
        v4f an[MT];
        #pragma unroll
        for (int i = 0; i < MT; ++i)
            an[i] = *(const v4f*)(Ap + (size_t)i * ArowStep + koff);

        #pragma unroll
        for (int i = 0; i < MT; ++i) {
            acc[i] = __builtin_amdgcn_wmma_f32_16x16x4_f32(
                false, lo2(ac[i]), false, lo2(bc), (short)0, acc[i], false, false);
            acc[i] = __builtin_amdgcn_wmma_f32_16x16x4_f32(
                false, hi2(ac[i]), false, hi2(bc), (short)0, acc[i], false, false);
        }
        bc = bn;
        #pragma unroll
        for (int i = 0; i < MT; ++i) ac[i] = an[i];
    }
    // epilogue chunk
    #pragma unroll
    for (int i = 0; i < MT; ++i) {
        acc[i] = __builtin_amdgcn_wmma_f32_16x16x4_f32(
            false, lo2(ac[i]), false, lo2(bc), (short)0, acc[i], false, false);
        acc[i] = __builtin_amdgcn_wmma_f32_16x16x4_f32(
            false, hi2(ac[i]), false, hi2(bc), (short)0, acc[i], false, false);
    }

    const float bs = bias ? bias[n0 + lr] : 0.0f;
    #pragma unroll
    for (int i = 0; i < MT; ++i)
        #pragma unroll
        for (int j = 0; j < 8; ++j)
            C[(size_t)(m0 + 16 * i + j + 8 * hl) * (size_t)ldc + (n0 + lr)] =
                acc[i][j] + bs;
}

// ------------------------------ init / copy --------------------------------
__global__ void init_kernel(const float* __restrict__ eh, float* __restrict__ h0,
                            int* __restrict__ tok)
{
    int i = blockIdx.x * blockDim.x + threadIdx.x;
    if (i < BDIM * HDIM) h0[i] = eh[i];
    if (i < BDIM) tok[i] = 0;                   // SOS = 0
}
__global__ void copy_kernel(const float* __restrict__ src, float* __restrict__ dst, int n)
{
    int i = blockIdx.x * blockDim.x + threadIdx.x;
    if (i < n) dst[i] = src[i];
}

// ------------------------------ attention ----------------------------------
// e[b,s] = Va . tanh(q[b] + UaK[b,s]) + Va_b      grid=(S,B), block=256
__global__ void attn_scores_kernel(const float* __restrict__ q,
                                   const float* __restrict__ UaK,
                                   const float* __restrict__ Va_w,
                                   const float* __restrict__ Va_b,
                                   float* __restrict__ e)
{
    const int s = blockIdx.x, b = blockIdx.y;
    const float* qb = q + (size_t)b * HDIM;
    const float* u  = UaK + ((size_t)b * SDIM + s) * HDIM;
    float acc = 0.f;
    for (int i = threadIdx.x; i < HDIM; i += blockDim.x)
        acc += Va_w[i] * tanhf(qb[i] + u[i]);
    __shared__ float red[8];
    const int lane = threadIdx.x & 31, wid = threadIdx.x >> 5;
    acc = wave_sum(acc);
    if (lane == 0) red[wid] = acc;
    __syncthreads();
    if (threadIdx.x == 0) {
        float t = 0.f;
        #pragma unroll
        for (int w = 0; w < 8; ++w) t += red[w];
        e[(size_t)b * SDIM + s] = t + Va_b[0];
    }
}

// softmax over S, written straight into the attentions output region.
// grid=B, block=128 (S=100 fits one pass)
__global__ void attn_softmax_kernel(const float* __restrict__ e,
                                    float* __restrict__ attn, int t, int L)
{
    const int b = blockIdx.x;
    const int s = threadIdx.x;
    const int lane = threadIdx.x & 31, wid = threadIdx.x >> 5;
    float v = (s < SDIM) ? e[(size_t)b * SDIM + s] : -3.4e38f;

    __shared__ float rmax[4];
    float m = wave_max(v);
    if (lane == 0) rmax[wid] = m;
    __syncthreads();
    m = fmaxf(fmaxf(rmax[0], rmax[1]), fmaxf(rmax[2], rmax[3]));

    float ex = (s < SDIM) ? __expf(v - m) : 0.f;
    __shared__ float rsum[4];
    float sm = wave_sum(ex);
    if (lane == 0) rsum[wid] = sm;
    __syncthreads();
    sm = rsum[0] + rsum[1] + rsum[2] + rsum[3];

    if (s < SDIM) attn[((size_t)b * L + t) * SDIM + s] = ex / sm;
}

// x[b, 0:H]   = emb[tok[b]]
// x[b, H:3H]  = sum_s w[b,s] * enc[b,s,:]         grid=(3H/256, B)
__global__ void ctx_embed_kernel(const float* __restrict__ attn,
                                 const float* __restrict__ enc,
                                 const float* __restrict__ emb,
                                 const int* __restrict__ tok,
                                 float* __restrict__ x, int t, int L)
{
    const int b = blockIdx.y;
    const int d = blockIdx.x * blockDim.x + threadIdx.x;     // < 3H
    float* xb = x + (size_t)b * 3 * HDIM;
    if (d < HDIM) {
        xb[d] = emb[(size_t)tok[b] * HDIM + d];
    } else {
        const int dd = d - HDIM;                             // < 2H
        const float* wb = attn + ((size_t)b * L + t) * SDIM;
        const float* eb = enc + (size_t)b * SDIM * 2 * HDIM + dd;
        float acc = 0.f;
        #pragma unroll 4
        for (int s2 = 0; s2 < SDIM; ++s2)
            acc += wb[s2] * eb[(size_t)s2 * 2 * HDIM];
        xb[d] = acc;
    }
}

// ------------------------------ GRU gating ---------------------------------
// torch gate order [r;z;n].  h_new = (1-z)*n + z*h
__global__ void gru_fuse_kernel(const float* __restrict__ gi,
                                const float* __restrict__ gh,
                                const float* __restrict__ h,
                                float* __restrict__ hn)
{
    const int i = blockIdx.x * blockDim.x + threadIdx.x;     // < B*H
    if (i >= BDIM * HDIM) return;
    const int b = i / HDIM, j = i % HDIM;
    const float* gib = gi + (size_t)b * 3 * HDIM;
    const float* ghb = gh + (size_t)b * 3 * HDIM;
    const float r = sigmoidf_(gib[j]            + ghb[j]);
    const float z = sigmoidf_(gib[HDIM + j]     + ghb[HDIM + j]);
    const float n = tanhf(    gib[2 * HDIM + j] + r * ghb[2 * HDIM + j]);
    hn[i] = (1.f - z) * n + z * h[i];
}

// ------------------------------ argmax (greedy) ----------------------------
// First-max semantics like jnp.argmax.  grid=B, block=256.
__global__ void argmax_kernel(const float* __restrict__ logits, long long ldc,
                              int* __restrict__ tok)
{
    const int b = blockIdx.x;
    const float* row = logits + (size_t)b * (size_t)ldc;
    float best = -3.4e38f; int bi = 0x7fffffff;
    for (int v = threadIdx.x; v < VDIM; v += blockDim.x) {
        float xv = row[v];
        if (xv > best) { best = xv; bi = v; }
    }
    __shared__ float rv[8]; __shared__ int ri[8];
    const int lane = threadIdx.x & 31, wid = threadIdx.x >> 5;
    wave_argmax(best, bi);
    if (lane == 0) { rv[wid] = best; ri[wid] = bi; }
    __syncthreads();
    if (threadIdx.x == 0) {
        #pragma unroll
        for (int w = 1; w < 8; ++w)
            if (rv[w] > best || (rv[w] == best && ri[w] < bi)) { best = rv[w]; bi = ri[w]; }
        tok[b] = bi;
    }
}

// ------------------------------ host-side ----------------------------------
// M must be a multiple of 64 (16*MT with MT=4); true for all call sites
// (B=64 and B*S=6400).
static inline void launch_gemm(const float* A, int lda, const float* Bt, int ldb,
                               const float* bias, float* C, long long ldc,
                               int M, int N, int K, hipStream_t s)
{
    dim3 grid((N / 16 + 3) / 4, M / 64);
    gemm_atbt_kernel<4><<<grid, 128, 0, s>>>(A, lda, Bt, ldb, bias, C, ldc, N, K);
}

extern "C" void kernel_launch(void* const* d_in, const int* in_sizes, int n_in,
                              void* d_out, int out_size, void* d_ws, size_t ws_size,
                              hipStream_t stream)
{
    (void)in_sizes; (void)n_in; (void)ws_size;
    const float* enc   = (const float*)d_in[0];   // [B,S,2H]
    const float* eh    = (const float*)d_in[1];   // [1,B,H]
    const float* emb   = (const float*)d_in[3];   // [V,H]
    const float* Wa_w  = (const float*)d_in[4];   // [H,H]
    const float* Wa_b  = (const float*)d_in[5];
    const float* Ua_w  = (const float*)d_in[6];   // [H,2H]
    const float* Ua_b  = (const float*)d_in[7];
    const float* Va_w  = (const float*)d_in[8];   // [1,H]
    const float* Va_b  = (const float*)d_in[9];
    const float* W_ih  = (const float*)d_in[10];  // [3H,3H]
    const float* W_hh  = (const float*)d_in[11];  // [3H,H]
    const float* b_ih  = (const float*)d_in[12];
    const float* b_hh  = (const float*)d_in[13];
    const float* W_out = (const float*)d_in[14];  // [V,H]
    const float* b_out = (const float*)d_in[15];

    // L from out_size = B*(L*(V+S) + H)   (== 50 for the reference)
    int L = (int)(((long long)out_size - (long long)BDIM * HDIM) /
                  ((long long)BDIM * (VDIM + SDIM)));
    if (L < 1) L = 1;

    // workspace layout (floats), total ~28 MB
    float* ws  = (float*)d_ws;
    size_t off = 0;
    float* UaK = ws + off; off += (size_t)BDIM * SDIM * HDIM;   // [B,S,H]
    float* q   = ws + off; off += (size_t)BDIM * HDIM;          // [B,H]
    float* e   = ws + off; off += (size_t)BDIM * SDIM;          // [B,S]
    float* x   = ws + off; off += (size_t)BDIM * 3 * HDIM;      // [B,3H]
    float* gi  = ws + off; off += (size_t)BDIM * 3 * HDIM;
    float* gh  = ws + off; off += (size_t)BDIM * 3 * HDIM;
    float* h0  = ws + off; off += (size_t)BDIM * HDIM;
    float* h1  = ws + off; off += (size_t)BDIM * HDIM;
    int*   tok = (int*)(ws + off);

    // output regions (all f32): [B,L,V] ++ [1,B,H] ++ [B,L,S]
    float* dec  = (float*)d_out;
    float* hT   = dec + (size_t)BDIM * L * VDIM;
    float* attn = hT + (size_t)BDIM * HDIM;

    // init: h0 = encoder_hidden[0], tok = 0
    init_kernel<<<(BDIM * HDIM + 255) / 256, 256, 0, stream>>>(eh, h0, tok);

    // loop-invariant UaK = enc @ Ua_w^T + Ua_b : [B*S, 2H] x [H, 2H]^T
    launch_gemm(enc, 2 * HDIM, Ua_w, 2 * HDIM, Ua_b,
                UaK, HDIM, BDIM * SDIM, HDIM, 2 * HDIM, stream);

    const long long ldcL = (long long)L * VDIM;
    for (int t = 0; t < L; ++t) {
        float* h  = (t & 1) ? h1 : h0;
        float* hn = (t & 1) ? h0 : h1;

        // q = h @ Wa_w^T + Wa_b
        launch_gemm(h, HDIM, Wa_w, HDIM, Wa_b, q, HDIM, BDIM, HDIM, HDIM, stream);
        // e[b,s] = Va . tanh(q + UaK) + Va_b ; softmax -> attentions output
        attn_scores_kernel<<<dim3(SDIM, BDIM), 256, 0, stream>>>(q, UaK, Va_w, Va_b, e);
        attn_softmax_kernel<<<BDIM, 128, 0, stream>>>(e, attn, t, L);
        // x = [emb[tok], ctx]
        ctx_embed_kernel<<<dim3(3 * HDIM / 256, BDIM), 256, 0, stream>>>(
            attn, enc, emb, tok, x, t, L);
        // GRU gate GEMMs
        launch_gemm(x, 3 * HDIM, W_ih, 3 * HDIM, b_ih, gi, 3 * HDIM,
                    BDIM, 3 * HDIM, 3 * HDIM, stream);
        launch_gemm(h, HDIM, W_hh, HDIM, b_hh, gh, 3 * HDIM,
                    BDIM, 3 * HDIM, HDIM, stream);
        gru_fuse_kernel<<<(BDIM * HDIM + 255) / 256, 256, 0, stream>>>(gi, gh, h, hn);
        // logits -> decoder_outputs[:, t, :]  (ldc = L*V interleaves steps)
        launch_gemm(hn, HDIM, W_out, HDIM, b_out, dec + (size_t)t * VDIM, ldcL,
                    BDIM, VDIM, HDIM, stream);
        // greedy feedback token
        argmax_kernel<<<BDIM, 256, 0, stream>>>(dec + (size_t)t * VDIM, ldcL, tok);
    }

    // final hidden state -> [1,B,H]
    const float* hfin = (L & 1) ? h1 : h0;
    copy_kernel<<<(BDIM * HDIM + 255) / 256, 256, 0, stream>>>(hfin, hT, BDIM * HDIM);
}